// YOLOXloss_75359496175835
// MI455X (gfx1250) — compile-verified
//
#include <hip/hip_runtime.h>
#include <cstdint>

// ---------------- problem constants (from reference) ----------------
#define B_   16
#define G_   100
#define A_   21504           // 128*128 + 64*64 + 32*32
#define L0N  16384
#define L1N  4096
#define L2N  1024
#define CRAD 2.5f

constexpr int    BA = B_ * A_;        // 344064
constexpr int    BG = B_ * G_;        // 1600
// ws layout (in floats)
constexpr size_t OFF_BB  = 0;                         // float4[BA]: cx,cy,w,h
constexpr size_t OFF_FO  = (size_t)4 * BA;            // float2[BA]: fg_anchor, softplus(-logit)
constexpr size_t OFF_LG  = (size_t)6 * BA;            // float [BA]: obj logit
constexpr size_t OFF_TC  = (size_t)7 * BA;            // thresh cost  [BG]
constexpr size_t OFF_TI  = (size_t)7 * BA + BG;       // thresh idx   [BG] (int)
constexpr size_t OFF_RV  = (size_t)7 * BA + 2 * BG;   // row valid    [BG]
constexpr size_t OFF_ACC = (size_t)7 * BA + 3 * BG;   // [nfg, iou_sum, bce_sum, gts]
// total ws need: (7*344064 + 3*1600 + 8) * 4B  ~= 9.7 MB

// ---------------- device helpers ----------------
__device__ __forceinline__ float softplusf(float x) {
    return fmaxf(x, 0.f) + log1pf(expf(-fabsf(x)));   // numerically stable
}

__device__ __forceinline__ void anchor_geom(int a, float& xg, float& yg, float& s) {
    if (a < L0N)            { s = 8.f;  xg = (float)(a & 127);            yg = (float)(a >> 7); }
    else if (a < L0N + L1N) { int t = a - L0N;        s = 16.f; xg = (float)(t & 63); yg = (float)(t >> 6); }
    else                    { int t = a - L0N - L1N;  s = 32.f; xg = (float)(t & 31); yg = (float)(t >> 5); }
}

__device__ __forceinline__ float pair_iou(float gx, float gy, float gw, float gh,
                                          float px, float py, float pw, float ph) {
    float tlx = fmaxf(gx - 0.5f * gw, px - 0.5f * pw);
    float tly = fmaxf(gy - 0.5f * gh, py - 0.5f * ph);
    float brx = fminf(gx + 0.5f * gw, px + 0.5f * pw);
    float bry = fminf(gy + 0.5f * gh, py + 0.5f * ph);
    bool  en  = (tlx < brx) && (tly < bry);
    float ai  = en ? (brx - tlx) * (bry - tly) : 0.f;
    return ai / (gw * gh + pw * ph - ai + 1e-16f);
}

__device__ __forceinline__ void in_tests(float gx, float gy, float gw, float gh,
                                         float xc, float yc, float r,
                                         bool& ib, bool& ic) {
    float b1 = xc - (gx - 0.5f * gw);
    float b2 = (gx + 0.5f * gw) - xc;
    float b3 = yc - (gy - 0.5f * gh);
    float b4 = (gy + 0.5f * gh) - yc;
    ib = fminf(fminf(b1, b2), fminf(b3, b4)) > 0.f;
    float c1 = xc - gx + r;
    float c2 = gx + r - xc;
    float c3 = yc - gy + r;
    float c4 = gy + r - yc;
    ic = fminf(fminf(c1, c2), fminf(c3, c4)) > 0.f;
}

// async-stage labels[b] (500 floats = 125 x b128) into LDS via CDNA5 async path
__device__ __forceinline__ void stage_labels_async(float* smL, const float* labels_b, int tid) {
    if (tid < 125) {
        unsigned ldsOff = (unsigned)(uintptr_t)(&smL[0]) + (unsigned)(tid * 16);
        asm volatile("global_load_async_to_lds_b128 %0, %1, %2 offset:0"
                     :: "v"(ldsOff), "v"(tid * 16), "s"(labels_b)
                     : "memory");
    }
    asm volatile("s_wait_asynccnt 0" ::: "memory");
}

// ---------------- kernel 0: zero accumulators ----------------
__global__ void k_zero(float* __restrict__ ws) {
    if (threadIdx.x < 8) ws[OFF_ACC + threadIdx.x] = 0.f;
}

// ---------------- kernel 1: decode + fg_anchor ----------------
__global__ __launch_bounds__(256) void k_decode_fg(
    const float* __restrict__ reg0, const float* __restrict__ obj0,
    const float* __restrict__ reg1, const float* __restrict__ obj1,
    const float* __restrict__ reg2, const float* __restrict__ obj2,
    const float* __restrict__ labels, float* __restrict__ ws)
{
    __shared__ float smL[G_ * 5];
    const int tid = threadIdx.x;
    const int b   = blockIdx.x / (A_ / 256);           // A_ divisible by 256
    const int a   = (blockIdx.x % (A_ / 256)) * 256 + tid;

    stage_labels_async(smL, labels + (size_t)b * (G_ * 5), tid);
    __syncthreads();

    // decode
    const float* regp; const float* objp; int HH, loc;
    if (a < L0N)            { regp = reg0 + (size_t)b * 4 * L0N; objp = obj0 + (size_t)b * L0N; HH = L0N; loc = a; }
    else if (a < L0N + L1N) { regp = reg1 + (size_t)b * 4 * L1N; objp = obj1 + (size_t)b * L1N; HH = L1N; loc = a - L0N; }
    else                    { regp = reg2 + (size_t)b * 4 * L2N; objp = obj2 + (size_t)b * L2N; HH = L2N; loc = a - L0N - L1N; }

    float xg, yg, s; anchor_geom(a, xg, yg, s);
    float t0 = regp[loc];
    float t1 = regp[HH + loc];
    float t2 = regp[2 * HH + loc];
    float t3 = regp[3 * HH + loc];
    float lg = objp[loc];

    const int base = b * A_ + a;
    float4 bb;
    bb.x = (t0 + xg) * s;          // cx
    bb.y = (t1 + yg) * s;          // cy
    bb.z = expf(t2) * s;           // w
    bb.w = expf(t3) * s;           // h
    *(float4*)&ws[OFF_BB + (size_t)4 * base] = bb;     // one global_store_b128
    ws[OFF_LG + base] = lg;

    // fg_anchor: any_g(in_boxes) | any_g(in_centers)  (ALL 100 rows, incl. zero rows)
    float xc = (xg + 0.5f) * s, yc = (yg + 0.5f) * s, r = CRAD * s;
    bool fg = false;
    #pragma unroll 4
    for (int g = 0; g < G_; ++g) {
        float gx = smL[g * 5 + 1], gy = smL[g * 5 + 2];
        float gw = smL[g * 5 + 3], gh = smL[g * 5 + 4];
        bool ib, ic; in_tests(gx, gy, gw, gh, xc, yc, r, ib, ic);
        fg = fg || ib || ic;
    }
    float2 fo; fo.x = fg ? 1.f : 0.f; fo.y = softplusf(-lg);
    *(float2*)&ws[OFF_FO + (size_t)2 * base] = fo;     // one global_store_b64
}

// ---------------- kernel 2: per-(b,g) row stats (one wave32 per row) ----------------
// Computes dyn_k from top-10 masked IoUs, then the dyn_k-th smallest (cost, idx)
// pair (stable-sort tie-break) as the matching threshold.
__global__ __launch_bounds__(32) void k_rowstats(
    const float* __restrict__ labels, float* __restrict__ ws)
{
    const int row  = blockIdx.x;
    const int b    = row / G_;
    const int lane = threadIdx.x;

    const float* L = labels + (size_t)row * 5;
    float l0 = L[0], gx = L[1], gy = L[2], gw = L[3], gh = L[4];
    bool valid = (l0 + gx + gy + gw + gh) > 0.f;

    float* thrC = ws + OFF_TC;
    int*   thrI = (int*)(ws + OFF_TI);
    float* rowV = ws + OFF_RV;
    float* acc  = ws + OFF_ACC;

    if (!valid) {
        if (lane == 0) { rowV[row] = 0.f; thrC[row] = 3.402823466e38f; thrI[row] = 0x7fffffff; }
        return;
    }

    const float* bbA = ws + OFF_BB + (size_t)4 * b * A_;
    const float* foA = ws + OFF_FO + (size_t)2 * b * A_;

    // per-lane: 10 smallest (cost, idx) lexicographic; 10 largest masked ious
    float mnC[10]; int mnI[10]; float mxI[10];
    #pragma unroll
    for (int k = 0; k < 10; ++k) { mnC[k] = 3.402823466e38f; mnI[k] = 0x7fffffff; mxI[k] = -1.f; }

    float gx0 = gx - 0.5f * gw, gx1 = gx + 0.5f * gw;
    float gy0 = gy - 0.5f * gh, gy1 = gy + 0.5f * gh;
    float garea = gw * gh;

    // per-level scan: level boundaries are multiples of 32 -> wave-uniform ranges,
    // so s / xmask / yshift are loop-invariant (no per-iteration level branch).
    auto scan = [&](int aBeg, int aEnd, float s, int xmask, int yshift, int lvlOff) {
        float r = CRAD * s;
        for (int a = aBeg + lane; a < aEnd; a += 32) {
            __builtin_prefetch(bbA + 4 * (a + 128), 0, 0);   // global_prefetch_b8
            float4 bb = *(const float4*)&bbA[(size_t)4 * a]; // global_load_b128
            float2 fo = *(const float2*)&foA[(size_t)2 * a]; // global_load_b64
            float px = bb.x, py = bb.y, pw = bb.z, ph = bb.w;

            // iou
            float tlx = fmaxf(gx0, px - 0.5f * pw);
            float tly = fmaxf(gy0, py - 0.5f * ph);
            float brx = fminf(gx1, px + 0.5f * pw);
            float bry = fminf(gy1, py + 0.5f * ph);
            bool  en  = (tlx < brx) && (tly < bry);
            float ai  = en ? (brx - tlx) * (bry - tly) : 0.f;
            float iou = ai / (garea + pw * ph - ai + 1e-16f);

            int loc = a - lvlOff;
            float xc = ((float)(loc & xmask) + 0.5f) * s;
            float yc = ((float)(loc >> yshift) + 0.5f) * s;
            bool ib = fminf(fminf(xc - gx0, gx1 - xc), fminf(yc - gy0, gy1 - yc)) > 0.f;
            bool ic = fminf(fminf(xc - gx + r, gx + r - xc),
                            fminf(yc - gy + r, gy + r - yc)) > 0.f;
            bool inb = ib && ic;
            bool vp  = fo.x > 0.5f;           // row valid; pair valid iff fg_anchor

            float cost = fo.y - 3.f * logf(iou + 1e-8f)
                       + (inb ? 0.f : 100000.f)
                       + (vp  ? 0.f : 10000000.f);
            float miou = vp ? iou : 0.f;

            // keep 10 smallest (cost, idx) — fully unrolled register insertion
            if (cost < mnC[9] || (cost == mnC[9] && a < mnI[9])) {
                float cc = cost; int ii = a;
                #pragma unroll
                for (int k = 0; k < 10; ++k) {
                    bool lt = (cc < mnC[k]) || (cc == mnC[k] && ii < mnI[k]);
                    float nc = lt ? cc : mnC[k];  int ni = lt ? ii : mnI[k];
                    float oc = lt ? mnC[k] : cc;  int oi = lt ? mnI[k] : ii;
                    mnC[k] = nc; mnI[k] = ni; cc = oc; ii = oi;
                }
            }
            // keep 10 largest masked ious
            if (miou > mxI[9]) {
                float vv = miou;
                #pragma unroll
                for (int k = 0; k < 10; ++k) {
                    float hi = fmaxf(vv, mxI[k]);
                    float lo = fminf(vv, mxI[k]);
                    mxI[k] = hi; vv = lo;
                }
            }
        }
    };
    scan(0,          L0N,        8.f, 127, 7, 0);
    scan(L0N,        L0N + L1N, 16.f,  63, 6, L0N);
    scan(L0N + L1N,  A_,        32.f,  31, 5, L0N + L1N);

    // wave merge 1: sum of 10 global-largest masked ious -> dyn_k
    float sum10 = 0.f;
    for (int rnd = 0; rnd < 10; ++rnd) {
        float cand = mxI[0];
        float m = cand;
        #pragma unroll
        for (int off = 16; off >= 1; off >>= 1) m = fmaxf(m, __shfl_xor(m, off, 32));
        sum10 += m;
        unsigned long long ball = __ballot(cand == m);
        int src = __ffsll(ball) - 1;
        if (lane == src) {
            #pragma unroll
            for (int k = 0; k < 9; ++k) mxI[k] = mxI[k + 1];
            mxI[9] = -1.f;
        }
    }
    int dynk = (int)sum10;            // trunc toward zero, like astype(int32)
    if (dynk < 1)  dynk = 1;
    if (dynk > 10) dynk = 10;

    // wave merge 2: extract 10 global-smallest (cost, idx); record the dyn_k-th
    float Tc = 0.f; int Ti = 0;
    for (int rnd = 0; rnd < 10; ++rnd) {
        float c0 = mnC[0]; int i0 = mnI[0];
        float mc = c0; int mi = i0;
        #pragma unroll
        for (int off = 16; off >= 1; off >>= 1) {
            float oc2 = __shfl_xor(mc, off, 32);
            int   oi2 = __shfl_xor(mi, off, 32);
            if (oc2 < mc || (oc2 == mc && oi2 < mi)) { mc = oc2; mi = oi2; }
        }
        if (rnd == dynk - 1) { Tc = mc; Ti = mi; }
        if (c0 == mc && i0 == mi) {   // unique owner pops
            #pragma unroll
            for (int k = 0; k < 9; ++k) { mnC[k] = mnC[k + 1]; mnI[k] = mnI[k + 1]; }
            mnC[9] = 3.402823466e38f; mnI[9] = 0x7fffffff;
        }
    }

    if (lane == 0) {
        thrC[row] = Tc; thrI[row] = Ti; rowV[row] = 1.f;
        atomicAdd(&acc[3], 1.f);       // num_gts
    }
}

// ---------------- kernel 3: per-anchor assignment + loss accumulation ----------------
__global__ __launch_bounds__(256) void k_assign(
    const float* __restrict__ labels, float* __restrict__ ws)
{
    __shared__ float smL[G_ * 5];
    __shared__ float smTc[G_];
    __shared__ int   smTi[G_];
    __shared__ float smV[G_];

    const int tid = threadIdx.x;
    const int b   = blockIdx.x / (A_ / 256);
    const int a   = (blockIdx.x % (A_ / 256)) * 256 + tid;

    stage_labels_async(smL, labels + (size_t)b * (G_ * 5), tid);

    const float* thrC = ws + OFF_TC;
    const int*   thrI = (const int*)(ws + OFF_TI);
    const float* rowV = ws + OFF_RV;
    for (int i = tid; i < G_; i += 256) {
        smTc[i] = thrC[b * G_ + i];
        smTi[i] = thrI[b * G_ + i];
        smV[i]  = rowV[b * G_ + i];
    }
    __syncthreads();

    const int base = b * A_ + a;
    float4 bb = *(const float4*)&ws[OFF_BB + (size_t)4 * base];
    float2 fo = *(const float2*)&ws[OFF_FO + (size_t)2 * base];
    float  lg = ws[OFF_LG + base];
    float px = bb.x, py = bb.y, pw = bb.z, ph = bb.w;
    float fga = fo.x, ocst = fo.y;

    float xg, yg, s; anchor_geom(a, xg, yg, s);
    float xc = (xg + 0.5f) * s, yc = (yg + 0.5f) * s, r = CRAD * s;

    float bestC = 3.402823466e38f; int bestG = 0;
    int nm = 0; int firstG = 0; bool haveFirst = false;

    for (int g = 0; g < G_; ++g) {
        float gx = smL[g * 5 + 1], gy = smL[g * 5 + 2];
        float gw = smL[g * 5 + 3], gh = smL[g * 5 + 4];
        bool valid = smV[g] > 0.5f;

        float iou = pair_iou(gx, gy, gw, gh, px, py, pw, ph);
        bool ib, ic; in_tests(gx, gy, gw, gh, xc, yc, r, ib, ic);
        bool inb = ib && ic;
        bool vp  = valid && (fga > 0.5f);

        float cost = ocst - 3.f * logf(iou + 1e-8f)
                   + (inb ? 0.f : 100000.f)
                   + (vp  ? 0.f : 10000000.f);

        if (cost < bestC) { bestC = cost; bestG = g; }   // first-index tie-break

        bool match = valid && ((cost < smTc[g]) || (cost == smTc[g] && a <= smTi[g]));
        if (match) { ++nm; if (!haveFirst) { haveFirst = true; firstG = g; } }
    }

    bool fg; int mg;
    if (nm > 1) { fg = smV[bestG] > 0.5f; mg = fg ? bestG : 0; }
    else        { fg = (nm == 1);          mg = firstG; }

    float fgf  = fg ? 1.f : 0.f;
    float liou = 0.f;
    if (fg) {
        float gx = smL[mg * 5 + 1], gy = smL[mg * 5 + 2];
        float gw = smL[mg * 5 + 3], gh = smL[mg * 5 + 4];
        float iou = pair_iou(gx, gy, gw, gh, px, py, pw, ph);
        liou = 1.f - iou * iou;
    }
    float bce = softplusf(lg) - lg * fgf;

    // wave32 reduction then one atomic per wave
    float vN = fgf, vI = liou, vB = bce;
    #pragma unroll
    for (int off = 16; off >= 1; off >>= 1) {
        vN += __shfl_down(vN, off, 32);
        vI += __shfl_down(vI, off, 32);
        vB += __shfl_down(vB, off, 32);
    }
    if ((tid & 31) == 0) {
        float* acc = ws + OFF_ACC;
        atomicAdd(&acc[0], vN);
        atomicAdd(&acc[1], vI);
        atomicAdd(&acc[2], vB);
    }
}

// ---------------- kernel 4: finalize scalars ----------------
__global__ void k_final(const float* __restrict__ ws, float* __restrict__ out) {
    const float* acc = ws + OFF_ACC;
    float nfg    = acc[0];
    float num_fg = fmaxf(nfg, 1.f);
    float li     = acc[1] / num_fg;
    float lo     = acc[2] / num_fg;
    float gts    = fmaxf(acc[3], 1.f);
    out[0] = 5.f * li + lo;
    out[1] = 5.f * li;
    out[2] = lo;
    out[3] = 0.f;
    out[4] = num_fg / gts;
}

// ---------------- host entry ----------------
extern "C" void kernel_launch(void* const* d_in, const int* in_sizes, int n_in,
                              void* d_out, int out_size, void* d_ws, size_t ws_size,
                              hipStream_t stream)
{
    // setup_inputs() dict order: reg0, obj0, reg1, obj1, reg2, obj2, labels
    const float* reg0   = (const float*)d_in[0];
    const float* obj0   = (const float*)d_in[1];
    const float* reg1   = (const float*)d_in[2];
    const float* obj1   = (const float*)d_in[3];
    const float* reg2   = (const float*)d_in[4];
    const float* obj2   = (const float*)d_in[5];
    const float* labels = (const float*)d_in[6];
    float* ws  = (float*)d_ws;      // needs ~9.7 MB
    float* out = (float*)d_out;

    k_zero     <<<1, 32, 0, stream>>>(ws);
    k_decode_fg<<<B_ * (A_ / 256), 256, 0, stream>>>(reg0, obj0, reg1, obj1, reg2, obj2, labels, ws);
    k_rowstats <<<BG, 32, 0, stream>>>(labels, ws);
    k_assign   <<<B_ * (A_ / 256), 256, 0, stream>>>(labels, ws);
    k_final    <<<1, 1, 0, stream>>>(ws, out);
}